// CLloss_47081431499216
// MI455X (gfx1250) — compile-verified
//
#include <hip/hip_runtime.h>
#include <math.h>
#include <stdint.h>

typedef float v2f __attribute__((ext_vector_type(2)));
typedef float v8f __attribute__((ext_vector_type(8)));

#define T_INV    10.0f   // 1 / T, T = 0.1
#define PD_EPS   1e-6
#define NORM_EPS 1e-12
#define MAX_DIM  2048    // reference embed dim

// ---------------------------------------------------------------------------
// Stage 1: for each 16-row tile, compute (dot with row0, row-sum, row-sumsq)
// using V_WMMA_F32_16X16X4_F32 as a multi-column reduction engine.
//   A (16x4)  = 16 embed rows, K-chunk of 4
//   B (4x16)  = col0: embed0 chunk (from LDS), col1: ones, cols 2..15: zero
//   C1 += A  x B  -> col0: dot_j partial, col1: sum_j partial
//   C2 += A2 x B  -> col1: sumsq_j partial   (A2 = A elementwise squared)
// 8 waves per block, each owning a disjoint K-slice; manual 4x unroll keeps
// 4 global loads in flight per wave. embed row 0 is staged into LDS once per
// block via the CDNA5 async global->LDS path.
// ---------------------------------------------------------------------------
__global__ __launch_bounds__(256) void cl_stage1(const float* __restrict__ embed,
                                                 float* __restrict__ rowstats,
                                                 int dim) {
  const int lane  = threadIdx.x & 31;
  const int wave  = threadIdx.x >> 5;
  const int m     = lane & 15;   // M index (A) / N index (B)
  const int khalf = lane >> 4;   // 0 -> K{0,1}, 1 -> K{2,3}
  const int row0  = blockIdx.x * 16;

  __shared__ float se0[MAX_DIM];               // embed row 0 (8 KB for dim=2048)

  // Async-copy embed[0][:] into LDS: 16-byte chunks, coalesced across threads.
  {
    const int nchunks = dim >> 2;              // float4 chunks
    for (int c = threadIdx.x; c < nchunks; c += blockDim.x) {
      unsigned lds_off = (unsigned)(uintptr_t)(&se0[c * 4]);
      unsigned long long gaddr = (unsigned long long)(uintptr_t)(embed + c * 4);
      asm volatile("global_load_async_to_lds_b128 %0, %1, off"
                   :: "v"(lds_off), "v"(gaddr) : "memory");
    }
    asm volatile("s_wait_asynccnt 0x0" ::: "memory");
  }
  __syncthreads();

  const int kslice = dim >> 3;                 // dim/8 columns per wave
  const int kbeg   = wave * kslice;
  const int kend   = kbeg + kslice;

  const float* rowp = embed + (size_t)(row0 + m) * dim + 2 * khalf;
  const float* e0l  = se0 + 2 * khalf;

  v8f c1 = {0.f, 0.f, 0.f, 0.f, 0.f, 0.f, 0.f, 0.f};
  v8f c2 = {0.f, 0.f, 0.f, 0.f, 0.f, 0.f, 0.f, 0.f};

  const bool  isN0 = (m == 0);
  const float bcon = (m == 1) ? 1.0f : 0.0f;   // ones column / zero columns

  for (int k = kbeg; k < kend; k += 16) {
    // Batch the global loads so >=4 are in flight before any wait.
    float2 av0 = *reinterpret_cast<const float2*>(rowp + k);
    float2 av1 = *reinterpret_cast<const float2*>(rowp + k + 4);
    float2 av2 = *reinterpret_cast<const float2*>(rowp + k + 8);
    float2 av3 = *reinterpret_cast<const float2*>(rowp + k + 12);
    // e0 chunks come from LDS (DScnt path, decoupled from LOADcnt).
    float2 bv0 = *reinterpret_cast<const float2*>(e0l + k);
    float2 bv1 = *reinterpret_cast<const float2*>(e0l + k + 4);
    float2 bv2 = *reinterpret_cast<const float2*>(e0l + k + 8);
    float2 bv3 = *reinterpret_cast<const float2*>(e0l + k + 12);

    v2f a, a2, b;
#define CL_STEP(AV, BV)                                                        \
    a.x  = AV.x;         a.y  = AV.y;                                          \
    a2.x = AV.x * AV.x;  a2.y = AV.y * AV.y;                                   \
    b.x  = isN0 ? BV.x : bcon;                                                 \
    b.y  = isN0 ? BV.y : bcon;                                                 \
    c1 = __builtin_amdgcn_wmma_f32_16x16x4_f32(false, a,  false, b, (short)0,  \
                                               c1, false, false);              \
    c2 = __builtin_amdgcn_wmma_f32_16x16x4_f32(false, a2, false, b, (short)0,  \
                                               c2, false, false);
    CL_STEP(av0, bv0)
    CL_STEP(av1, bv1)
    CL_STEP(av2, bv2)
    CL_STEP(av3, bv3)
#undef CL_STEP
  }

  // C/D layout: VGPR r -> rows (M=r, lanes 0..15) and (M=r+8, lanes 16..31),
  // lane%16 = N.  We need N=0 (dot) and N=1 (sum / sumsq).
  __shared__ float part[8][16][3];
  #pragma unroll
  for (int r = 0; r < 8; ++r) {
    if (lane == 0)  { part[wave][r][0]     = c1[r]; }
    if (lane == 1)  { part[wave][r][1]     = c1[r]; part[wave][r][2]     = c2[r]; }
    if (lane == 16) { part[wave][r + 8][0] = c1[r]; }
    if (lane == 17) { part[wave][r + 8][1] = c1[r]; part[wave][r + 8][2] = c2[r]; }
  }
  __syncthreads();

  const int t = threadIdx.x;
  if (t < 48) {
    const int rr = t / 3, cc = t % 3;
    float s = 0.f;
    #pragma unroll
    for (int w = 0; w < 8; ++w) s += part[w][rr][cc];
    rowstats[(size_t)(row0 + rr) * 3 + cc] = s;
  }
}

// ---------------------------------------------------------------------------
// Stage 2: per-row loss terms + deterministic block-level partial sums (f64).
//   S1 = sum c_j, S2 = sum exp(-d_j/T), S3 = sum c_j * (-d_j/T)
// ---------------------------------------------------------------------------
__global__ __launch_bounds__(256) void cl_stage2(const float* __restrict__ rowstats,
                                                 const int* __restrict__ labels,
                                                 double* __restrict__ blockpart,
                                                 int nrows, int dim, int ldim) {
  const float sq0  = rowstats[2];
  const float sum0 = rowstats[1];
  const float n0   = fmaxf(sqrtf(sq0), (float)NORM_EPS);
  const float A2   = sq0 / (n0 * n0);
  const float SA   = sum0 / n0;
  const float epsc = (float)(2.0 * PD_EPS);
  const float dterm = (float)((double)dim * PD_EPS * PD_EPS);

  double s1 = 0.0, s2 = 0.0, s3 = 0.0;

  for (int j = 1 + blockIdx.x * blockDim.x + threadIdx.x; j < nrows;
       j += gridDim.x * blockDim.x) {
    const float dotj = rowstats[(size_t)j * 3 + 0];
    const float sumj = rowstats[(size_t)j * 3 + 1];
    const float sqj  = rowstats[(size_t)j * 3 + 2];

    const float nj = fmaxf(sqrtf(sqj), (float)NORM_EPS);
    const float B2 = sqj / (nj * nj);
    const float AB = dotj / (n0 * nj);
    const float SB = sumj / nj;

    float dsq = A2 + B2 - 2.0f * AB + epsc * (SA - SB) + dterm;
    const float d  = sqrtf(fmaxf(dsq, 0.0f));
    const float ls = -d * T_INV;                  // log_sim_j

    int c = 0;
    const int* labj = labels + (size_t)j * ldim;
    for (int q = 0; q < ldim; ++q) c += labj[q] * labels[q];  // labels[0] cached

    const double cd = (double)c;
    s1 += cd;
    s2 += exp((double)ls);
    s3 += cd * (double)ls;
  }

  __shared__ double sh[3][256];
  sh[0][threadIdx.x] = s1; sh[1][threadIdx.x] = s2; sh[2][threadIdx.x] = s3;
  __syncthreads();
  for (int off = 128; off > 0; off >>= 1) {
    if (threadIdx.x < off) {
      sh[0][threadIdx.x] += sh[0][threadIdx.x + off];
      sh[1][threadIdx.x] += sh[1][threadIdx.x + off];
      sh[2][threadIdx.x] += sh[2][threadIdx.x + off];
    }
    __syncthreads();
  }
  if (threadIdx.x == 0) {
    blockpart[blockIdx.x * 3 + 0] = sh[0][0];
    blockpart[blockIdx.x * 3 + 1] = sh[1][0];
    blockpart[blockIdx.x * 3 + 2] = sh[2][0];
  }
}

// ---------------------------------------------------------------------------
// Stage 3: fold block partials into the scalar loss.
// ---------------------------------------------------------------------------
__global__ void cl_stage3(const double* __restrict__ blockpart, int nblocks,
                          float* __restrict__ out, int nrows) {
  if (threadIdx.x == 0 && blockIdx.x == 0) {
    double S1 = 0.0, S2 = 0.0, S3 = 0.0;
    for (int b = 0; b < nblocks; ++b) {
      S1 += blockpart[b * 3 + 0];
      S2 += blockpart[b * 3 + 1];
      S3 += blockpart[b * 3 + 2];
    }
    const double Ci = 1e-12 + S1;
    const double Ei = 1e-12 + S2;
    const double Li = -(S3 - S1 * log(Ei)) / Ci;
    out[0] = (float)(Li / (double)nrows);
  }
}

extern "C" void kernel_launch(void* const* d_in, const int* in_sizes, int n_in,
                              void* d_out, int out_size, void* d_ws, size_t ws_size,
                              hipStream_t stream) {
  const float* embed  = (const float*)d_in[0];
  const int*   labels = (const int*)d_in[1];

  const int LDIM  = 20;                       // labels second dim (reference)
  const int nrows = in_sizes[1] / LDIM;       // 16384
  const int dim   = in_sizes[0] / nrows;      // 2048

  // Workspace layout: [rowstats: nrows*3 f32][blockpart: NB2*3 f64]
  float* rowstats = (float*)d_ws;
  size_t rsBytes  = (size_t)nrows * 3 * sizeof(float);
  double* blockpart = (double*)((char*)d_ws + ((rsBytes + 15) & ~(size_t)15));

  const int NB2 = 64;

  cl_stage1<<<nrows / 16, 256, 0, stream>>>(embed, rowstats, dim);
  cl_stage2<<<NB2, 256, 0, stream>>>(rowstats, labels, blockpart, nrows, dim, LDIM);
  cl_stage3<<<1, 32, 0, stream>>>(blockpart, NB2, (float*)d_out, nrows);
}